// ProposalTargetLayer_offset_12515534701034
// MI455X (gfx1250) — compile-verified
//
#include <hip/hip_runtime.h>
#include <cstdint>

// Problem constants (from reference setup_inputs)
constexpr int BATCH   = 16;
constexpr int NPRIOR  = 32768;
constexpr int NTRUTH  = 128;
constexpr int TOPK    = 4;
constexpr int CHUNK   = 2048;                 // priors per LDS tile (32 KB of float4)
constexpr int NCHUNK  = NPRIOR / CHUNK;       // 16
constexpr int TPB1    = 256;                  // 8 waves (wave32)
constexpr int WAVES1  = TPB1 / 32;
constexpr int TSPLIT  = 4;                    // truth-split for occupancy
constexpr int TRUTHS_PER_BLOCK = NTRUTH / TSPLIT;          // 32
constexpr int TRUTHS_PER_WAVE  = TRUTHS_PER_BLOCK / WAVES1; // 4
constexpr int NCAND   = NCHUNK * TOPK;        // 64 candidates per (b,g)
constexpr float VAR_XY = 0.1f;
constexpr float VAR_WH = 0.2f;

// ---------------------------------------------------------------------------
// Sortable key: high 32 bits = float bits of IoU (IoU >= 0 so unsigned-
// monotone), low 32 bits = ~prior_index (equal IoU -> lower index is larger
// key). Bigger key == better candidate == jax.lax.top_k order.
// ---------------------------------------------------------------------------
__device__ __forceinline__ uint64_t make_key(float iou, int idx) {
    return ((uint64_t)__float_as_uint(iou) << 32) | (uint32_t)(~idx);
}
__device__ __forceinline__ int key_index(uint64_t k) {
    return (int)(~(uint32_t)k);
}

// branchless compare-exchange: a keeps the larger key (1x v_cmp_gt_u64)
__device__ __forceinline__ void cswap(uint64_t& a, uint64_t& b) {
    const bool c = b > a;
    const uint64_t mx = c ? b : a;
    const uint64_t mn = c ? a : b;
    a = mx; b = mn;
}

// branchless insert into sorted-descending top-4
__device__ __forceinline__ void ins_top4(uint64_t key, uint64_t& k0,
                                         uint64_t& k1, uint64_t& k2,
                                         uint64_t& k3) {
    k3 = (key > k3) ? key : k3;
    cswap(k2, k3);
    cswap(k1, k2);
    cswap(k0, k1);
}

// ordering-only IoU -> sortable key (fast v_rcp_f32 instead of IEEE divide)
__device__ __forceinline__ uint64_t iou_key(const float4 pr, const float4 t,
                                            float area_t, int gid) {
    const float ltx = fmaxf(t.x, pr.x);
    const float lty = fmaxf(t.y, pr.y);
    const float rbx = fminf(t.z, pr.z);
    const float rby = fminf(t.w, pr.w);
    const float w   = fmaxf(rbx - ltx, 0.0f);
    const float h   = fmaxf(rby - lty, 0.0f);
    const float inter  = w * h;
    const float area_p = (pr.z - pr.x) * (pr.w - pr.y);
    const float iou = inter * __builtin_amdgcn_rcpf(area_t + area_p - inter);
    return make_key(iou, gid);
}

__device__ __forceinline__ uint64_t shfl_xor_u64(uint64_t k, int mask) {
    uint32_t lo = (uint32_t)k;
    uint32_t hi = (uint32_t)(k >> 32);
    lo = __shfl_xor((int)lo, mask, 32);
    hi = __shfl_xor((int)hi, mask, 32);
    return ((uint64_t)hi << 32) | lo;
}

// ---------------------------------------------------------------------------
// Kernel 1: per (image, truth-split, prior-chunk) partial top-4.
// Prior tile staged into LDS via CDNA5 async global->LDS DMA; each HBM byte
// is read once (repeat fetches hit the 192MB L2) and reused out of LDS.
// ---------------------------------------------------------------------------
__global__ __launch_bounds__(TPB1)
void iou_partial_topk(const float* __restrict__ rois,    // [B,2,P,4]
                      const float* __restrict__ targets, // [B,G,4]
                      uint64_t* __restrict__ pkeys) {    // [B,G,NCHUNK,TOPK]
    const int chunk = blockIdx.x;
    const int tsp   = blockIdx.y;
    const int b     = blockIdx.z;
    const int tid   = threadIdx.x;
    const int lane  = tid & 31;
    const int wave  = tid >> 5;

    __shared__ float4 tile[CHUNK];    // 32 KB prior tile

    // ---- async stage: global (defaults = rois[b,1]) tile -> LDS ----------
    {
        const float* gsrc =
            rois + ((size_t)(b * 2 + 1) * NPRIOR + (size_t)chunk * CHUNK) * 4;
        const uint64_t sbase = (uint64_t)(uintptr_t)gsrc;
#pragma unroll
        for (int r = 0; r < CHUNK / TPB1; ++r) {
            const int      pi   = r * TPB1 + tid;       // prior within tile
            const uint32_t voff = (uint32_t)pi * 16u;   // byte offset in global
            // low 32 bits of a flat LDS pointer == LDS byte address
            const uint32_t ldsa = (uint32_t)(uintptr_t)(&tile[pi]);
            asm volatile("global_load_async_to_lds_b128 %0, %1, %2"
                         :: "v"(ldsa), "v"(voff), "s"(sbase)
                         : "memory");
        }
        asm volatile("s_wait_asynccnt 0" ::: "memory");
    }
    __syncthreads();

    const int gbase = chunk * CHUNK;

    // ---- each wave owns TRUTHS_PER_WAVE truths ----------------------------
#pragma unroll
    for (int tg = 0; tg < TRUTHS_PER_WAVE; ++tg) {
        const int g = tsp * TRUTHS_PER_BLOCK + wave * TRUTHS_PER_WAVE + tg;
        const float4 t =
            *(const float4*)(targets + ((size_t)b * NTRUTH + g) * 4);
        const float area_t = (t.z - t.x) * (t.w - t.y);

        // two independent top-4 chains -> 2x ILP on the dependent cswap chain
        uint64_t a0 = 0, a1 = 0, a2 = 0, a3 = 0;
        uint64_t c0 = 0, c1 = 0, c2 = 0, c3 = 0;

#pragma unroll 4
        for (int p = lane; p < CHUNK; p += 64) {
            const float4 prA = tile[p];        // ds_load_b128
            const float4 prB = tile[p + 32];   // ds_load_b128
            const uint64_t ka = iou_key(prA, t, area_t, gbase + p);
            const uint64_t kb = iou_key(prB, t, area_t, gbase + p + 32);
            ins_top4(ka, a0, a1, a2, a3);
            ins_top4(kb, c0, c1, c2, c3);
        }

        // merge chain C into chain A (bitonic merge of sorted-4 lists)
        cswap(a0, c3);
        cswap(a1, c2);
        cswap(a2, c1);
        cswap(a3, c0);
        cswap(a0, a2);
        cswap(a1, a3);
        cswap(a0, a1);
        cswap(a2, a3);

        // ---- wave32 butterfly merge of sorted-4 key lists -----------------
#pragma unroll
        for (int off = 16; off >= 1; off >>= 1) {
            uint64_t u0 = shfl_xor_u64(a0, off);
            uint64_t u1 = shfl_xor_u64(a1, off);
            uint64_t u2 = shfl_xor_u64(a2, off);
            uint64_t u3 = shfl_xor_u64(a3, off);
            // Bitonic merge of [a0..a3 desc, u3..u0 asc]: 8 compare-exchanges
            cswap(a0, u3);
            cswap(a1, u2);
            cswap(a2, u1);
            cswap(a3, u0);
            cswap(a0, a2);
            cswap(a1, a3);
            cswap(a0, a1);
            cswap(a2, a3);
        }

        if (lane == 0) {
            const size_t o =
                (((size_t)b * NTRUTH + g) * NCHUNK + chunk) * TOPK;
            pkeys[o + 0] = a0;
            pkeys[o + 1] = a1;
            pkeys[o + 2] = a2;
            pkeys[o + 3] = a3;
        }
    }
}

// ---------------------------------------------------------------------------
// Kernel 2: final top-4 across chunks + gather matched priors + encode.
// One block per (b, g). Output layout: out[b, g*4 + k, 4].
// ---------------------------------------------------------------------------
__global__ __launch_bounds__(64)
void final_topk_encode(const float* __restrict__ rois,
                       const float* __restrict__ targets,
                       const uint64_t* __restrict__ pkeys,
                       float* __restrict__ out) {
    const int bg = blockIdx.x;
    const int b  = bg / NTRUTH;
    const int t  = threadIdx.x;

    __shared__ uint64_t sk[NCAND];

    sk[t] = pkeys[(size_t)bg * NCAND + t];
    __syncthreads();

    if (t == 0) {
        int sel[TOPK];
#pragma unroll
        for (int k = 0; k < TOPK; ++k) {
            uint64_t best = 0;
            int arg = 0;
            for (int c = 0; c < NCAND; ++c) {
                const uint64_t kc = sk[c];
                const bool g = kc > best;     // single u64 compare
                best = g ? kc : best;
                arg  = g ? c : arg;
            }
            sel[k] = key_index(best);
            sk[arg] = 0;                      // exclude for next rank
        }

        const float4 tt = *(const float4*)(targets + (size_t)bg * 4);
        const float g_cx = (tt.x + tt.z) * 0.5f;
        const float g_cy = (tt.y + tt.w) * 0.5f;
        const float g_w  = tt.z - tt.x;
        const float g_h  = tt.w - tt.y;

#pragma unroll
        for (int k = 0; k < TOPK; ++k) {
            const int pi = sel[k];
            const float4 m = *(const float4*)(
                rois + ((size_t)(b * 2 + 1) * NPRIOR + (size_t)pi) * 4);
            const float p_cx = (m.x + m.z) * 0.5f;
            const float p_cy = (m.y + m.w) * 0.5f;
            const float p_w  = m.z - m.x;
            const float p_h  = m.w - m.y;
            float4 o;
            o.x = (g_cx - p_cx) / (VAR_XY * p_w);
            o.y = (g_cy - p_cy) / (VAR_XY * p_h);
            o.z = logf(g_w / p_w) / VAR_WH;
            o.w = logf(g_h / p_h) / VAR_WH;
            *(float4*)(out + ((size_t)bg * TOPK + k) * 4) = o;
        }
    }
}

// ---------------------------------------------------------------------------
extern "C" void kernel_launch(void* const* d_in, const int* in_sizes, int n_in,
                              void* d_out, int out_size, void* d_ws, size_t ws_size,
                              hipStream_t stream) {
    const float* rois    = (const float*)d_in[0];  // [16,2,32768,4] f32
    const float* targets = (const float*)d_in[1];  // [16,128,4] f32
    // d_in[2] = expand_num (== 4, baked into TOPK)

    uint64_t* pkeys = (uint64_t*)d_ws;             // 1 MB of scratch

    dim3 grid1(NCHUNK, TSPLIT, BATCH);             // 1024 workgroups
    iou_partial_topk<<<grid1, dim3(TPB1), 0, stream>>>(rois, targets, pkeys);

    final_topk_encode<<<dim3(BATCH * NTRUTH), dim3(64), 0, stream>>>(
        rois, targets, pkeys, (float*)d_out);
}